// GNNPolicy_21930103013553
// MI455X (gfx1250) — compile-verified
//
#include <hip/hip_runtime.h>
#include <hip/hip_bf16.h>
#include <cstdint>

// ---------------------------------------------------------------------------
// GNNPolicy forward for MI455X (gfx1250, wave32).
// fp32 end-to-end (memory-bound problem; precision is free), all 64-wide
// GEMMs on V_WMMA_F32_16X16X4_F32. Edge pipeline fully fused:
//   msg = projR[dst] + (prenorm(ef)@We_emb+b_emb)@We_conv + projL[src]
//   out = relu(msg*scale)@Wf + bf ; atomic segment-sum into agg[dst]
// ---------------------------------------------------------------------------

#define NCONS  100000
#define NVARS  200000
#define NEDGES 1000000
#define TSTRIDE 66   // padded LDS transpose-tile stride (even, for b64 loads)

typedef float v2f __attribute__((ext_vector_type(2)));
typedef float v8f __attribute__((ext_vector_type(8)));

__device__ __forceinline__ v8f wmma4(v2f a, v2f b, v8f c) {
  // D(16x16,f32) += A(16x4,f32) * B(4x16,f32)
  return __builtin_amdgcn_wmma_f32_16x16x4_f32(false, a, false, b, (short)0, c,
                                               false, false);
}

// Pack row-major W[K][64] (zero-padded to Kpad rows) into LDS in the WMMA
// B-operand layout: lane L (h=L>=16, c=L&15) reads, for k-step ks,
// the pair { W[4ks+2h][n], W[4ks+2h+1][n] } as one b64.
__device__ __forceinline__ void pack_w(float* lds, const float* W, int K, int Kpad) {
  for (int i = threadIdx.x; i < Kpad * 64; i += blockDim.x) {
    int k = i >> 6, n = i & 63;
    float v = (k < K) ? W[k * 64 + n] : 0.0f;
    int ks = k >> 2, h = (k >> 1) & 1, j = k & 1;
    lds[((ks * 2 + h) * 64 + n) * 2 + j] = v;
  }
}

__device__ __forceinline__ void zero_acc(v8f acc[4]) {
#pragma unroll
  for (int nt = 0; nt < 4; ++nt)
#pragma unroll
    for (int r = 0; r < 8; ++r) acc[nt][r] = 0.0f;
}

// D[16 x 64] += A[16 x 4*KS] * W[4*KS x 64].  A held per-lane in a[2*KS]
// (a[2k+j] = A[lane&15][4k + 2h + j]), W pre-packed in LDS (pack_w).
template <int KS>
__device__ __forceinline__ void gemm_acc(const float* a, const float* ldsW, v8f acc[4]) {
  const int h = (threadIdx.x >> 4) & 1;
  const int c = threadIdx.x & 15;
#pragma unroll
  for (int nt = 0; nt < 4; ++nt) {
    v8f d = acc[nt];
#pragma unroll
    for (int k = 0; k < KS; ++k) {
      v2f av; av[0] = a[2 * k]; av[1] = a[2 * k + 1];
      float2 bq = ((const float2*)ldsW)[(k * 2 + h) * 64 + nt * 16 + c];
      v2f bv; bv[0] = bq.x; bv[1] = bq.y;
      d = wmma4(av, bv, d);
    }
    acc[nt] = d;
  }
}

// D-layout (lane holds column nt*16+c, rows r+8h) -> A-layout via wave-private
// LDS tile. Same-wave DS ops are in-order on CDNA5; explicit dscnt wait added.
__device__ __forceinline__ void d_to_a(const v8f acc[4], float* tile, float* a) {
  const int h = (threadIdx.x >> 4) & 1;
  const int c = threadIdx.x & 15;
#pragma unroll
  for (int nt = 0; nt < 4; ++nt)
#pragma unroll
    for (int r = 0; r < 8; ++r)
      tile[(r + 8 * h) * TSTRIDE + nt * 16 + c] = acc[nt][r];
  asm volatile("s_wait_dscnt 0" ::: "memory");
#pragma unroll
  for (int k = 0; k < 16; ++k) {
    float2 q = *(const float2*)&tile[c * TSTRIDE + 4 * k + 2 * h];
    a[2 * k] = q.x;
    a[2 * k + 1] = q.y;
  }
}

__device__ __forceinline__ void add_bias_d(v8f acc[4], const float* b) {
  const int c = threadIdx.x & 15;
#pragma unroll
  for (int nt = 0; nt < 4; ++nt) {
    float bb = b[nt * 16 + c];
#pragma unroll
    for (int r = 0; r < 8; ++r) acc[nt][r] += bb;
  }
}

__device__ __forceinline__ void relu_d(v8f acc[4]) {
#pragma unroll
  for (int nt = 0; nt < 4; ++nt)
#pragma unroll
    for (int r = 0; r < 8; ++r) acc[nt][r] = fmaxf(acc[nt][r], 0.0f);
}

__device__ __forceinline__ void store_d(const v8f acc[4], float* out, int rowbase, int nrows) {
  const int h = (threadIdx.x >> 4) & 1;
  const int c = threadIdx.x & 15;
#pragma unroll
  for (int nt = 0; nt < 4; ++nt)
#pragma unroll
    for (int r = 0; r < 8; ++r) {
      int row = rowbase + r + 8 * h;
      if (row < nrows) out[(size_t)row * 64 + nt * 16 + c] = acc[nt][r];
    }
}

__device__ __forceinline__ void load_a64(float* a, const float* X, int row) {
  const int h = (threadIdx.x >> 4) & 1;
#pragma unroll
  for (int k = 0; k < 16; ++k) {
    float2 q = *(const float2*)&X[(size_t)row * 64 + 4 * k + 2 * h];
    a[2 * k] = q.x;
    a[2 * k + 1] = q.y;
  }
}

// ---------------------------------------------------------------------------
// Embedding: prenorm -> Linear(KIN,64) -> ReLU -> Linear(64,64) -> ReLU
// ---------------------------------------------------------------------------
#define EMB_BLOCK 128
#define EMB_WAVES 4

template <int KIN, int KS1, bool PRES>
__global__ __launch_bounds__(EMB_BLOCK) void emb_kernel(
    const float* __restrict__ X, const float* __restrict__ avg,
    const float* __restrict__ scale, const unsigned char* __restrict__ pmask,
    const float* __restrict__ W1, const float* __restrict__ b1,
    const float* __restrict__ W2, const float* __restrict__ b2,
    float* __restrict__ OUT, int nrows) {
  __shared__ float ldsW1[KS1 * 256];
  __shared__ float ldsW2[4096];
  __shared__ float tiles[EMB_WAVES][16 * TSTRIDE];
  pack_w(ldsW1, W1, KIN, KS1 * 4);
  pack_w(ldsW2, W2, 64, 64);
  __syncthreads();
  const int wave = threadIdx.x >> 5;
  const int rowbase = (blockIdx.x * EMB_WAVES + wave) * 16;
  if (rowbase >= nrows) return;
  const int h = (threadIdx.x >> 4) & 1;
  const int c = threadIdx.x & 15;
  const int row = min(rowbase + c, nrows - 1);

  float a1[2 * KS1];
#pragma unroll
  for (int k = 0; k < KS1; ++k) {
#pragma unroll
    for (int j = 0; j < 2; ++j) {
      int pos = 4 * k + 2 * h + j;
      float y = 0.0f;
      if (pos < KIN) {
        float v = X[(size_t)row * KIN + pos];
        y = (v - avg[pos]) * scale[pos];
        if (PRES && pmask[pos]) y = v;  // preserve_features passthrough
      }
      a1[2 * k + j] = y;
    }
  }
  v8f acc[4]; zero_acc(acc);
  gemm_acc<KS1>(a1, ldsW1, acc);
  add_bias_d(acc, b1); relu_d(acc);
  float a2[32];
  d_to_a(acc, tiles[wave], a2);
  v8f acc2[4]; zero_acc(acc2);
  gemm_acc<16>(a2, ldsW2, acc2);
  add_bias_d(acc2, b2); relu_d(acc2);
  store_d(acc2, OUT, rowbase, nrows);
}

// ---------------------------------------------------------------------------
// Node projection: OUT = X @ W (+ bias)
// ---------------------------------------------------------------------------
#define PROJ_BLOCK 256
#define PROJ_WAVES 8

__global__ __launch_bounds__(PROJ_BLOCK) void proj_kernel(
    const float* __restrict__ X, const float* __restrict__ W,
    const float* __restrict__ bias, float* __restrict__ OUT, int nrows) {
  __shared__ float ldsW[4096];
  pack_w(ldsW, W, 64, 64);
  __syncthreads();
  const int wave = threadIdx.x >> 5;
  const int rowbase = (blockIdx.x * PROJ_WAVES + wave) * 16;
  if (rowbase >= nrows) return;
  const int c = threadIdx.x & 15;
  const int row = min(rowbase + c, nrows - 1);
  float a[32];
  load_a64(a, X, row);
  v8f acc[4]; zero_acc(acc);
  gemm_acc<16>(a, ldsW, acc);
  if (bias) add_bias_d(acc, bias);
  store_d(acc, OUT, rowbase, nrows);
}

// ---------------------------------------------------------------------------
// Fused edge pipeline (one wave = 16 edges):
//   E = prenorm(ef)@We_emb + b_emb            (built in registers, K=2)
//   M = E@We (WMMA) ; transpose via LDS
//   T = relu((M + projR[dst] + projL[src]) * scale_final)
//   O = T@Wf (WMMA) + bf ; atomicAdd agg[dst] += O
// ---------------------------------------------------------------------------
#define EDGE_BLOCK 128
#define EDGE_WAVES 4

__global__ __launch_bounds__(EDGE_BLOCK) void edge_kernel(
    const float* __restrict__ efeat, const int* __restrict__ dst_idx,
    const int* __restrict__ src_idx, const float* __restrict__ e_avg,
    const float* __restrict__ e_scale, const float* __restrict__ eW,
    const float* __restrict__ eb, const float* __restrict__ projR,
    const float* __restrict__ projL, const float* __restrict__ We,
    const float* __restrict__ sfin, const float* __restrict__ Wf,
    const float* __restrict__ bf, float* __restrict__ agg, int nE) {
  __shared__ float ldsWe[4096];
  __shared__ float ldsWf[4096];
  __shared__ float sWp[192];  // [0:64)=eW row0, [64:128)=eW row1, [128:192)=eb
  __shared__ float tiles[EDGE_WAVES][16 * TSTRIDE];
  pack_w(ldsWe, We, 64, 64);
  pack_w(ldsWf, Wf, 64, 64);
  for (int i = threadIdx.x; i < 192; i += blockDim.x)
    sWp[i] = (i < 128) ? eW[i] : eb[i - 128];
  __syncthreads();

  const int wave = threadIdx.x >> 5;
  const int base = (blockIdx.x * EDGE_WAVES + wave) * 16;
  if (base >= nE) return;
  const int h = (threadIdx.x >> 4) & 1;
  const int c = threadIdx.x & 15;
  const int eid = min(base + c, nE - 1);
  const int dl = dst_idx[eid];
  const int sl = src_idx[eid];
  const float z0 = (efeat[(size_t)eid * 2 + 0] - e_avg[0]) * e_scale[0];
  const float z1 = (efeat[(size_t)eid * 2 + 1] - e_avg[1]) * e_scale[1];

  // Edge embedding row, directly in A-layout (K=2 GEMV in registers)
  float a[32];
#pragma unroll
  for (int k = 0; k < 16; ++k) {
    int pos = 4 * k + 2 * h;
    float2 w0 = *(const float2*)&sWp[pos];
    float2 w1 = *(const float2*)&sWp[64 + pos];
    float2 bb = *(const float2*)&sWp[128 + pos];
    a[2 * k]     = z0 * w0.x + z1 * w1.x + bb.x;
    a[2 * k + 1] = z0 * w0.y + z1 * w1.y + bb.y;
  }

  v8f acc[4]; zero_acc(acc);
  gemm_acc<16>(a, ldsWe, acc);       // msg = E @ We   (D-layout)
  d_to_a(acc, tiles[wave], a);       // back to A-layout

  // + projR[dst] + projL[src], * scale_final, ReLU   (gathers hit L2)
#pragma unroll
  for (int k = 0; k < 16; ++k) {
    int pos = 4 * k + 2 * h;
    float2 pr = *(const float2*)&projR[(size_t)dl * 64 + pos];
    float2 pl = *(const float2*)&projL[(size_t)sl * 64 + pos];
    float2 sf = *(const float2*)&sfin[pos];
    a[2 * k]     = fmaxf((a[2 * k] + pr.x + pl.x) * sf.x, 0.0f);
    a[2 * k + 1] = fmaxf((a[2 * k + 1] + pr.y + pl.y) * sf.y, 0.0f);
  }

  v8f acc2[4]; zero_acc(acc2);
  gemm_acc<16>(a, ldsWf, acc2);      // O = T @ Wf
  add_bias_d(acc2, bf);

  // segment-sum: coalesced fp32 atomics into L2-resident agg
#pragma unroll
  for (int r = 0; r < 8; ++r) {
    int eg = base + r + 8 * h;
    if (eg < nE) {
      int drow = dst_idx[eg];
#pragma unroll
      for (int nt = 0; nt < 4; ++nt)
        unsafeAtomicAdd(&agg[(size_t)drow * 64 + nt * 16 + c], acc2[nt][r]);
    }
  }
}

// ---------------------------------------------------------------------------
// Output MLP: relu(concat(agg*scale_post, right) @ Wo1 + bo1) @ Wo2 + bo2
// Implemented as one K=128 WMMA GEMM (A = [agg*sp | right]) + transpose + K=64.
// ---------------------------------------------------------------------------
#define OUT_BLOCK 64
#define OUT_WAVES 2

__global__ __launch_bounds__(OUT_BLOCK) void outmlp_kernel(
    const float* __restrict__ agg, const float* __restrict__ spost,
    const float* __restrict__ base_feat, const float* __restrict__ Wo1,
    const float* __restrict__ bo1, const float* __restrict__ Wo2,
    const float* __restrict__ bo2, float* __restrict__ OUT, int nrows) {
  __shared__ float ldsW1[8192];  // 128x64 packed
  __shared__ float ldsW2[4096];
  __shared__ float tiles[OUT_WAVES][16 * TSTRIDE];
  pack_w(ldsW1, Wo1, 128, 128);
  pack_w(ldsW2, Wo2, 64, 64);
  __syncthreads();
  const int wave = threadIdx.x >> 5;
  const int rowbase = (blockIdx.x * OUT_WAVES + wave) * 16;
  if (rowbase >= nrows) return;
  const int h = (threadIdx.x >> 4) & 1;
  const int c = threadIdx.x & 15;
  const int row = min(rowbase + c, nrows - 1);

  float a[64];
#pragma unroll
  for (int k = 0; k < 16; ++k) {
    int pos = 4 * k + 2 * h;
    float2 g = *(const float2*)&agg[(size_t)row * 64 + pos];
    float2 s = *(const float2*)&spost[pos];
    a[2 * k]     = g.x * s.x;
    a[2 * k + 1] = g.y * s.y;
    float2 bfe = *(const float2*)&base_feat[(size_t)row * 64 + pos];
    a[32 + 2 * k]     = bfe.x;
    a[32 + 2 * k + 1] = bfe.y;
  }
  v8f acc[4]; zero_acc(acc);
  gemm_acc<32>(a, ldsW1, acc);
  add_bias_d(acc, bo1); relu_d(acc);
  float a2[32];
  d_to_a(acc, tiles[wave], a2);
  v8f acc2[4]; zero_acc(acc2);
  gemm_acc<16>(a2, ldsW2, acc2);
  add_bias_d(acc2, bo2);
  store_d(acc2, OUT, rowbase, nrows);
}

// ---------------------------------------------------------------------------
// Head: out = relu(v@W1+b1) @ W2   (64->64->1), one wave per row
// ---------------------------------------------------------------------------
__global__ __launch_bounds__(256) void head_kernel(
    const float* __restrict__ V, const float* __restrict__ W1,
    const float* __restrict__ b1, const float* __restrict__ W2,
    float* __restrict__ out, int nrows) {
  int wid = (blockIdx.x * blockDim.x + threadIdx.x) >> 5;
  int lane = threadIdx.x & 31;
  if (wid >= nrows) return;
  const float* row = V + (size_t)wid * 64;
  float h0 = 0.f, h1 = 0.f;
#pragma unroll 8
  for (int k = 0; k < 64; ++k) {
    float x = row[k];
    h0 = fmaf(x, W1[k * 64 + lane], h0);
    h1 = fmaf(x, W1[k * 64 + lane + 32], h1);
  }
  float p = fmaxf(h0 + b1[lane], 0.f) * W2[lane] +
            fmaxf(h1 + b1[lane + 32], 0.f) * W2[lane + 32];
#pragma unroll
  for (int m = 16; m > 0; m >>= 1) p += __shfl_xor(p, m, 32);
  if (lane == 0) out[wid] = p;
}

// ---------------------------------------------------------------------------
extern "C" void kernel_launch(void* const* d_in, const int* in_sizes, int n_in,
                              void* d_out, int out_size, void* d_ws, size_t ws_size,
                              hipStream_t stream) {
  // Inputs flattened in setup_inputs() insertion order (recursively).
  const float* cons    = (const float*)d_in[0];
  const int*   eidx    = (const int*)d_in[1];
  const float* efeat   = (const float*)d_in[2];
  const float* var     = (const float*)d_in[3];
  const unsigned char* pmask = (const unsigned char*)d_in[4];
  const float* c_avg = (const float*)d_in[5],  *c_scale = (const float*)d_in[6];
  const float* c_W1  = (const float*)d_in[7],  *c_b1    = (const float*)d_in[8];
  const float* c_W2  = (const float*)d_in[9],  *c_b2    = (const float*)d_in[10];
  const float* e_avg = (const float*)d_in[11], *e_scale = (const float*)d_in[12];
  const float* e_W   = (const float*)d_in[13], *e_b     = (const float*)d_in[14];
  const float* v_avg = (const float*)d_in[15], *v_scale = (const float*)d_in[16];
  const float* v_W1  = (const float*)d_in[17], *v_b1    = (const float*)d_in[18];
  const float* v_W2  = (const float*)d_in[19], *v_b2    = (const float*)d_in[20];
  // conv params order: Wl,bl,We,Wr,scale_final,Wf,bf,scale_post,Wo1,bo1,Wo2,bo2
  const float* const* cv1 = (const float* const*)&d_in[21];
  const float* const* cv2 = (const float* const*)&d_in[33];
  const float* o_W1 = (const float*)d_in[45];
  const float* o_b1 = (const float*)d_in[46];
  const float* o_W2 = (const float*)d_in[47];
  (void)in_sizes; (void)n_in; (void)out_size; (void)ws_size;

  const int* ci = eidx;
  const int* vi = eidx + NEDGES;

  float* ws = (float*)d_ws;
  size_t off = 0;
  float* c_emb = ws + off; off += (size_t)NCONS * 64;
  float* v_emb = ws + off; off += (size_t)NVARS * 64;
  float* projR = ws + off; off += (size_t)NVARS * 64;
  float* projL = ws + off; off += (size_t)NVARS * 64;
  float* agg   = ws + off; off += (size_t)NVARS * 64;
  float* c_new = ws + off; off += (size_t)NCONS * 64;
  float* v_new = ws + off;

  auto nblocks = [](int rows, int wavesPerBlock) {
    int t = (rows + 15) / 16;
    return (t + wavesPerBlock - 1) / wavesPerBlock;
  };

  // Embeddings
  emb_kernel<10, 3, false><<<nblocks(NCONS, EMB_WAVES), EMB_BLOCK, 0, stream>>>(
      cons, c_avg, c_scale, nullptr, c_W1, c_b1, c_W2, c_b2, c_emb, NCONS);
  emb_kernel<104, 26, true><<<nblocks(NVARS, EMB_WAVES), EMB_BLOCK, 0, stream>>>(
      var, v_avg, v_scale, pmask, v_W1, v_b1, v_W2, v_b2, v_emb, NVARS);

  // Conv1: v -> c  (dst=ci, src=vi; right=c_emb, left=v_emb)
  proj_kernel<<<nblocks(NCONS, PROJ_WAVES), PROJ_BLOCK, 0, stream>>>(
      c_emb, cv1[0], cv1[1], projR, NCONS);
  proj_kernel<<<nblocks(NVARS, PROJ_WAVES), PROJ_BLOCK, 0, stream>>>(
      v_emb, cv1[3], nullptr, projL, NVARS);
  hipMemsetAsync(agg, 0, (size_t)NCONS * 64 * sizeof(float), stream);
  edge_kernel<<<nblocks(NEDGES, EDGE_WAVES), EDGE_BLOCK, 0, stream>>>(
      efeat, ci, vi, e_avg, e_scale, e_W, e_b, projR, projL,
      cv1[2], cv1[4], cv1[5], cv1[6], agg, NEDGES);
  outmlp_kernel<<<nblocks(NCONS, OUT_WAVES), OUT_BLOCK, 0, stream>>>(
      agg, cv1[7], c_emb, cv1[8], cv1[9], cv1[10], cv1[11], c_new, NCONS);

  // Conv2: c -> v  (dst=vi, src=ci; right=v_emb, left=c_new)
  proj_kernel<<<nblocks(NVARS, PROJ_WAVES), PROJ_BLOCK, 0, stream>>>(
      v_emb, cv2[0], cv2[1], projR, NVARS);
  proj_kernel<<<nblocks(NCONS, PROJ_WAVES), PROJ_BLOCK, 0, stream>>>(
      c_new, cv2[3], nullptr, projL, NCONS);
  hipMemsetAsync(agg, 0, (size_t)NVARS * 64 * sizeof(float), stream);
  edge_kernel<<<nblocks(NEDGES, EDGE_WAVES), EDGE_BLOCK, 0, stream>>>(
      efeat, vi, ci, e_avg, e_scale, e_W, e_b, projR, projL,
      cv2[2], cv2[4], cv2[5], cv2[6], agg, NEDGES);
  outmlp_kernel<<<nblocks(NVARS, OUT_WAVES), OUT_BLOCK, 0, stream>>>(
      agg, cv2[7], v_emb, cv2[8], cv2[9], cv2[10], cv2[11], v_new, NVARS);

  // Head
  head_kernel<<<(NVARS * 32 + 255) / 256, 256, 0, stream>>>(
      v_new, o_W1, o_b1, o_W2, (float*)d_out, NVARS);
}